// TransformerBlockQuantum_65481071401958
// MI455X (gfx1250) — compile-verified
//
#include <hip/hip_runtime.h>
#include <math.h>

typedef __attribute__((ext_vector_type(2))) float v2f;
typedef __attribute__((ext_vector_type(4))) float v4f;
typedef __attribute__((ext_vector_type(8))) float v8f;
typedef __attribute__((ext_vector_type(4))) unsigned int u32x4;
typedef __attribute__((ext_vector_type(4))) int i32x4;
typedef __attribute__((ext_vector_type(8))) int i32x8;

#define BB  4
#define SS  1024
#define DD  512
#define HH  64
#define DKK 8
#define FFD 2048
#define NQD 8
#define MM  (BB*SS)   // 4096 token rows

#if __has_builtin(__builtin_amdgcn_tensor_load_to_lds) && __has_builtin(__builtin_amdgcn_s_wait_tensorcnt)
#define HAVE_TDM 1
#endif

// K-chunk staged per double-buffer step; LDS row stride padded 32->36 dwords
#define KCH   32
#define LROW  36
#define ASZ   (128 * LROW)     // A tile floats (128 rows)
#define WSZ   (64  * LROW)     // W tile floats (64 rows)

#ifdef HAVE_TDM
// ---------------------------------------------------------------------------
// Issue one 2-D TDM tile load: tile = KCH columns x tile_rows rows of a
// row-major [dim1 x K] fp32 matrix starting at gaddr, into LDS at lds_byte.
// D# packing per CDNA5 ISA. Padding: after every
// 32 dwords (interval enc 4) insert 4 dwords (amount enc 3) -> row stride 144B.
// ---------------------------------------------------------------------------
__device__ inline void tdm_load_tile(unsigned int lds_byte, const float* gaddr,
                                     int K, int dim1, int tile_rows)
{
    unsigned long long ga = (unsigned long long)(uintptr_t)gaddr;
    u32x4 g0;
    g0[0] = 1u;                                   // count=1, user mode, no gather
    g0[1] = lds_byte;                             // lds_addr (bytes)
    g0[2] = (unsigned int)ga;                     // global_addr[31:0]
    g0[3] = (unsigned int)((ga >> 32) & 0x01FFFFFFu) | 0x80000000u; // addr[56:32] | type=2

    i32x8 g1;
    g1[0] = (2 << 16)            // data_size = 4 bytes
          | (1 << 20)            // pad_enable
          | (4 << 22)            // pad_interval enc: 32 dwords
          | (3 << 25);           // pad_amount  enc: 4 dwords
    g1[1] = (int)(((unsigned int)K & 0xFFFFu) << 16);                 // tensor_dim0 lo16
    g1[2] = (int)((((unsigned int)K >> 16) & 0xFFFFu)
          | (((unsigned int)dim1 & 0xFFFFu) << 16));                  // dim0 hi | dim1 lo
    g1[3] = (int)((((unsigned int)dim1 >> 16) & 0xFFFFu)
          | ((unsigned int)KCH << 16));                               // dim1 hi | tile_dim0
    g1[4] = tile_rows;                                                // tile_dim1 (tile_dim2=0)
    g1[5] = K;                                                        // tensor_dim0_stride lo32
    g1[6] = 0;
    g1[7] = 0;

    i32x4 g2 = {0, 0, 0, 0};
    i32x4 g3 = {0, 0, 0, 0};
#if defined(__clang_major__) && (__clang_major__ >= 23)
    i32x8 g4 = {0, 0, 0, 0, 0, 0, 0, 0};
    __builtin_amdgcn_tensor_load_to_lds(g0, g1, g2, g3, g4, 0);
#else
    __builtin_amdgcn_tensor_load_to_lds(g0, g1, g2, g3, 0);
#endif
}
#endif

// ---------------------------------------------------------------------------
// fp32 WMMA GEMM:  C[M x N] = A[M x K] @ W[N x K]^T + bias[N] (+ R)
// V_WMMA_F32_16X16X4_F32; block = 256 thr = 8 waves; block tile 128x64;
// wave tile 16x64 (4 accumulators). A/W K-chunks staged in double-buffered
// LDS by the Tensor Data Mover (wave 0 issues, TENSORcnt + barrier sync);
// cooperative-load fallback if the TDM builtin is unavailable.
// Fragment layouts (CDNA5 ISA, 16x16x4 f32):
//   A: lane l(0..15)=row, lane>>4 selects K{0,1}|{2,3};  B mirrors with l=col.
//   C/D: VGPR r -> rows r / r+8 split at lane 16, col = lane&15.
// skip_gate: result provably unused when *skip_gate == 1 -> uniform early-out.
// ---------------------------------------------------------------------------
__global__ __launch_bounds__(256) void gemm_wmma_f32(
    const float* __restrict__ A, const float* __restrict__ W,
    const float* __restrict__ bias, const float* __restrict__ R,
    float* __restrict__ C, int N, int K, const float* skip_gate)
{
    if (skip_gate && *skip_gate == 1.0f) return;

    __shared__ float Alds[2][ASZ];
    __shared__ float Wlds[2][WSZ];

    const int wave = threadIdx.x >> 5;
    const int lane = threadIdx.x & 31;
    const int half = lane >> 4;
    const int l    = lane & 15;
    const int row0 = blockIdx.y * 128;
    const int col0 = blockIdx.x * 64;
    const int nchunks = K / KCH;

#ifdef HAVE_TDM
    const unsigned int aBase = (unsigned int)(uintptr_t)&Alds[0][0];
    const unsigned int wBase = (unsigned int)(uintptr_t)&Wlds[0][0];
#endif

    auto stage = [&](int buf, int k0) {
#ifdef HAVE_TDM
        if (wave == 0) {
            __builtin_prefetch(A + (size_t)row0 * K + k0 + 2 * KCH, 0, 2);
            tdm_load_tile(aBase + (unsigned int)buf * ASZ * 4,
                          A + (size_t)row0 * K + k0, K, MM, 128);
            tdm_load_tile(wBase + (unsigned int)buf * WSZ * 4,
                          W + (size_t)col0 * K + k0, K, N, 64);
        }
#else
        // cooperative fill: A 128x32 (8 v4f/row), W 64x32
        for (int i = threadIdx.x; i < 128 * 8; i += 256) {
            int r = i >> 3, c4 = (i & 7) << 2;
            __builtin_prefetch(A + (size_t)(row0 + r) * K + k0 + c4 + 2 * KCH, 0, 2);
            *(v4f*)&Alds[buf][r * LROW + c4] =
                *(const v4f*)(A + (size_t)(row0 + r) * K + k0 + c4);
        }
        for (int i = threadIdx.x; i < 64 * 8; i += 256) {
            int r = i >> 3, c4 = (i & 7) << 2;
            *(v4f*)&Wlds[buf][r * LROW + c4] =
                *(const v4f*)(W + (size_t)(col0 + r) * K + k0 + c4);
        }
#endif
    };

    v8f acc[4] = {};
    stage(0, 0);
    for (int c = 0; c < nchunks; ++c) {
        const int cur = c & 1;
        if (c + 1 < nchunks) {
            stage(cur ^ 1, (c + 1) * KCH);       // async prefetch of next chunk
#ifdef HAVE_TDM
            if (wave == 0) __builtin_amdgcn_s_wait_tensorcnt(2);  // chunk c done
#endif
        }
#ifdef HAVE_TDM
        else if (wave == 0) { __builtin_amdgcn_s_wait_tensorcnt(0); }
#endif
        __syncthreads();                          // chunk c visible to all waves

        const float* aB = &Alds[cur][(wave * 16 + l) * LROW + 2 * half];
        const float* wB = &Wlds[cur][l * LROW + 2 * half];
        #pragma unroll
        for (int kk = 0; kk < KCH; kk += 4) {
            v2f a = *(const v2f*)(aB + kk);
            #pragma unroll
            for (int t = 0; t < 4; ++t) {
                v2f b = *(const v2f*)(wB + t * (16 * LROW) + kk);
                acc[t] = __builtin_amdgcn_wmma_f32_16x16x4_f32(
                    false, a, false, b, (short)0, acc[t], false, false);
            }
        }
        __syncthreads();                          // all reads done before re-stage
    }

    #pragma unroll
    for (int r = 0; r < 8; ++r) {
        const int row = row0 + wave * 16 + r + 8 * half;
        const size_t base = (size_t)row * N + col0 + l;
        #pragma unroll
        for (int t = 0; t < 4; ++t) {
            float o = acc[t][r] + bias[col0 + t * 16 + l];
            if (R) o += R[base + t * 16];
            C[base + t * 16] = o;
        }
    }
}

// ---------------------------------------------------------------------------
// Attention: quantum path (cos/cumprod on q) always; classical flash attention
// (online softmax, DK=8) only when gate != 1 (uniform, exact skip).
// One thread = one query row; block = 128 rows of one (b,h).
// ---------------------------------------------------------------------------
__global__ __launch_bounds__(128) void attn_kernel(
    const float* __restrict__ q, const float* __restrict__ k,
    const float* __restrict__ v, const float* __restrict__ theta,
    const float* __restrict__ gate_p, float* __restrict__ out)
{
    const int s = blockIdx.x * 128 + threadIdx.x;
    const int h = blockIdx.y;
    const int b = blockIdx.z;
    const float gate = *gate_p;

    const size_t rowbase = ((size_t)(b * SS + s)) * DD + h * DKK;
    float qv[DKK];
    #pragma unroll
    for (int d = 0; d < DKK; ++d) qv[d] = q[rowbase + d];

    float c[DKK];
    #pragma unroll
    for (int d = 0; d < DKK; ++d) c[d] = cosf(qv[d] + theta[d]);
    float z0 = 1.f;
    #pragma unroll
    for (int d = 1; d < DKK; ++d) z0 *= c[d];
    float quantum[DKK];
    quantum[0] = z0;
    float cum = c[0];
    #pragma unroll
    for (int d = 1; d < DKK; ++d) { cum *= c[d]; quantum[d] = cum; }

    float cls[DKK] = {0.f,0.f,0.f,0.f,0.f,0.f,0.f,0.f};
    if (gate != 1.0f) {
        __shared__ float kt[128 * DKK];
        __shared__ float vt[128 * DKK];
        float m = -1e30f, lsum = 0.f;
        const float scale = 0.35355339059327373f;   // 1/sqrt(DK)
        for (int tb = 0; tb < SS; tb += 128) {
            const size_t kb = ((size_t)(b * SS + tb + threadIdx.x)) * DD + h * DKK;
            #pragma unroll
            for (int d = 0; d < DKK; ++d) {
                kt[threadIdx.x * DKK + d] = k[kb + d];
                vt[threadIdx.x * DKK + d] = v[kb + d];
            }
            __syncthreads();
            for (int tt = 0; tt < 128; ++tt) {
                float sc = 0.f;
                #pragma unroll
                for (int d = 0; d < DKK; ++d) sc += qv[d] * kt[tt * DKK + d];
                sc *= scale;
                const float mnew = fmaxf(m, sc);
                const float corr = expf(m - mnew);
                const float p    = expf(sc - mnew);
                lsum = lsum * corr + p;
                #pragma unroll
                for (int d = 0; d < DKK; ++d)
                    cls[d] = cls[d] * corr + p * vt[tt * DKK + d];
                m = mnew;
            }
            __syncthreads();
        }
        const float inv = 1.0f / lsum;
        #pragma unroll
        for (int d = 0; d < DKK; ++d) cls[d] *= inv;
    }

    const float og = 1.0f - gate;
    #pragma unroll
    for (int d = 0; d < DKK; ++d)
        out[rowbase + d] = gate * quantum[d] + og * cls[d];
}

// ---------------------------------------------------------------------------
// LayerNorm over D=512, one wave32 per row (8 rows / 256-thread block).
// ---------------------------------------------------------------------------
__global__ __launch_bounds__(256) void ln_kernel(
    const float* __restrict__ in, const float* __restrict__ w,
    const float* __restrict__ bwt, float* __restrict__ out)
{
    const int wave = threadIdx.x >> 5, lane = threadIdx.x & 31;
    const int row = blockIdx.x * 8 + wave;
    const float* x = in + (size_t)row * DD;
    float s = 0.f, s2 = 0.f;
    for (int i = lane; i < DD; i += 32) { float t = x[i]; s += t; s2 += t * t; }
    #pragma unroll
    for (int off = 16; off > 0; off >>= 1) {
        s  += __shfl_xor(s,  off, 32);
        s2 += __shfl_xor(s2, off, 32);
    }
    const float mean = s * (1.0f / DD);
    const float var  = s2 * (1.0f / DD) - mean * mean;
    const float inv  = rsqrtf(var + 1e-5f);
    float* o = out + (size_t)row * DD;
    for (int i = lane; i < DD; i += 32)
        o[i] = (x[i] - mean) * inv * w[i] + bwt[i];
}

// ---------------------------------------------------------------------------
// FFN stage 1 (K=8, VALU): h = gate*relu(cos(xs)cos(phi)@W1^T+b1)
//                            + (1-gate)*relu(xs@W1^T+b1)
// Blending here lets stage 2 be one GEMM (b2 folds linearly).
// ---------------------------------------------------------------------------
__global__ __launch_bounds__(256) void ffn1_kernel(
    const float* __restrict__ x1, const float* __restrict__ W1,
    const float* __restrict__ b1, const float* __restrict__ phi,
    const float* __restrict__ gate_p, float* __restrict__ h)
{
    const int row = blockIdx.x;
    const float gate = *gate_p;
    const float og = 1.0f - gate;
    __shared__ float xs[NQD], qm[NQD];
    if (threadIdx.x < NQD) {
        float vv = x1[(size_t)row * DD + threadIdx.x];
        xs[threadIdx.x] = vv;
        qm[threadIdx.x] = cosf(vv) * cosf(phi[threadIdx.x]);
    }
    __syncthreads();
    float lxs[NQD], lqm[NQD];
    #pragma unroll
    for (int d = 0; d < NQD; ++d) { lxs[d] = xs[d]; lqm[d] = qm[d]; }
    for (int j = threadIdx.x; j < FFD; j += 256) {
        const float* wj = W1 + (size_t)j * NQD;
        float aq = b1[j], ac = aq;
        #pragma unroll
        for (int d = 0; d < NQD; ++d) { aq += lqm[d] * wj[d]; ac += lxs[d] * wj[d]; }
        h[(size_t)row * FFD + j] = gate * fmaxf(aq, 0.f) + og * fmaxf(ac, 0.f);
    }
}

// ---------------------------------------------------------------------------
extern "C" void kernel_launch(void* const* d_in, const int* in_sizes, int n_in,
                              void* d_out, int out_size, void* d_ws, size_t ws_size,
                              hipStream_t stream)
{
    const float* x      = (const float*)d_in[0];
    const float* Wq     = (const float*)d_in[1];
    const float* bq     = (const float*)d_in[2];
    const float* Wk     = (const float*)d_in[3];
    const float* bk     = (const float*)d_in[4];
    const float* Wv     = (const float*)d_in[5];
    const float* bv     = (const float*)d_in[6];
    const float* theta  = (const float*)d_in[7];
    const float* gate_a = (const float*)d_in[8];
    const float* Wo     = (const float*)d_in[9];
    const float* bo     = (const float*)d_in[10];
    const float* ln1w   = (const float*)d_in[11];
    const float* ln1b   = (const float*)d_in[12];
    const float* W1     = (const float*)d_in[13];
    const float* b1     = (const float*)d_in[14];
    const float* W2     = (const float*)d_in[15];
    const float* b2     = (const float*)d_in[16];
    const float* phi    = (const float*)d_in[17];
    const float* gate_f = (const float*)d_in[18];
    const float* ln2w   = (const float*)d_in[19];
    const float* ln2b   = (const float*)d_in[20];
    float* out = (float*)d_out;

    // workspace (floats), lifetime-aliased; peak 6*MD floats ~= 50 MB
    float* ws = (float*)d_ws;
    const size_t MD = (size_t)MM * DD;   // 2,097,152
    float* qb   = ws;                    // [0   , 1MD)
    float* kb   = ws + MD;               // [1MD , 2MD)
    float* vb   = ws + 2 * MD;           // [2MD , 3MD)
    float* atb  = ws + 3 * MD;           // [3MD , 4MD)  attn mix
    float* x1p  = ws;                    // reuse q   (dead after attn)
    float* x1   = ws + MD;               // reuse k
    float* hmix = ws + 2 * MD;           // [2MD , 6MD) reuse v+attn
    float* x2p  = ws;                    // reuse x1p

    const dim3 ggrid(DD / 64, MM / 128); // (8, 32)

    // Q always; K/V provably unused when gate_attn == 1 (skipped on-device)
    gemm_wmma_f32<<<ggrid, 256, 0, stream>>>(x, Wq, bq, nullptr, qb, DD, DD, nullptr);
    gemm_wmma_f32<<<ggrid, 256, 0, stream>>>(x, Wk, bk, nullptr, kb, DD, DD, gate_a);
    gemm_wmma_f32<<<ggrid, 256, 0, stream>>>(x, Wv, bv, nullptr, vb, DD, DD, gate_a);

    attn_kernel<<<dim3(SS / 128, HH, BB), 128, 0, stream>>>(qb, kb, vb, theta, gate_a, atb);

    // x1_pre = attn_mix @ Wo^T + bo + x ; then LN1
    gemm_wmma_f32<<<ggrid, 256, 0, stream>>>(atb, Wo, bo, x, x1p, DD, DD, nullptr);
    ln_kernel<<<MM / 8, 256, 0, stream>>>(x1p, ln1w, ln1b, x1);

    // FFN: blended hidden, then single big GEMM with residual; then LN2 -> out
    ffn1_kernel<<<MM, 256, 0, stream>>>(x1, W1, b1, phi, gate_f, hmix);
    gemm_wmma_f32<<<ggrid, 256, 0, stream>>>(hmix, W2, b2, x1, x2p, DD, FFD, nullptr);
    ln_kernel<<<MM / 8, 256, 0, stream>>>(x2p, ln2w, ln2b, out);
}